// Ridge_33749853012535
// MI455X (gfx1250) — compile-verified
//
#include <hip/hip_runtime.h>

// Growing-window ridge via blocked RLS (exact Sherman-Morrison recursion),
// WMMA-accelerated on gfx1250.
//
// Per batch b (256 independent batches, one wave32 workgroup each):
//   P = (1/lam) I  (32x32, LDS), w = 0 (32, LDS)
//   for each block of T=16 points X (16x32), y (16):
//     V = X * P                       -- WMMA f32 16x16x4 (16 instrs)
//     S = V * X^T                     -- WMMA (8 instrs)
//     v0 = X * w ; C = 0 ; g = 0
//     for m = 0..15 (sequential, compressed 16-dim space):
//       a  = e_m - C * S[:,m]
//       d  = 1 + S[:,m].a
//       pred = v0[m] + S[:,m].g       -> output (exact RLS prediction)
//       eps = y[m] - pred
//       g += a*eps/d ; C += a a^T / d
//     W = C * V                       -- WMMA (8 instrs)
//     P -= V^T * W                    -- WMMA (16 instrs)
//     w += V^T * g
// This reproduces ridge fit on [0,i) predicting point i exactly (pred_0 == 0).

typedef __attribute__((ext_vector_type(2))) float v2f;
typedef __attribute__((ext_vector_type(8))) float v8f;

#define BB 256
#define NN 256
#define DDIM 32
#define TBLK 16
#define RIDGE_LAM 1.0f

__launch_bounds__(32)
__global__ void ridge_rls_wmma(const float* __restrict__ data,
                               const float* __restrict__ targets,
                               float* __restrict__ out) {
  __shared__ float Psh[DDIM * DDIM];   // 32x32 inverse-Gram (ridge) matrix
  __shared__ float Vsh[TBLK * DDIM];   // 16x32  V = X P
  __shared__ float Xsh[TBLK * DDIM];   // 16x32  block of data rows
  __shared__ float Wsh[TBLK * DDIM];   // 16x32  W = C V
  __shared__ float Ssh[TBLK * TBLK];   // 16x16  S = X P X^T
  __shared__ float Csh[TBLK * TBLK];   // 16x16  compressed rank accumulator
  __shared__ float wsh[DDIM];          // 32 ridge weights
  __shared__ float ybs[TBLK];
  __shared__ float v0s[TBLK];
  __shared__ float avs[TBLK];
  __shared__ float gvs[TBLK];

  const int b    = blockIdx.x;
  const int lane = threadIdx.x;
  const int m    = lane & 15;          // fragment row (A) / col (B/C/D)
  const int g    = lane >> 4;          // lane half
  const int n    = m;                  // B/C/D column index
  const int k2   = 2 * g;              // K offset of this lane half

  const float* Xg = data    + (size_t)b * NN * DDIM;
  const float* Yg = targets + (size_t)b * NN;
  float*       Og = out     + (size_t)b * NN;

  // init P = I/lam, w = 0
  for (int idx = lane; idx < DDIM * DDIM; idx += 32)
    Psh[idx] = ((idx / DDIM) == (idx % DDIM)) ? (1.0f / RIDGE_LAM) : 0.0f;
  wsh[lane] = 0.0f;
  __syncthreads();

  for (int blk = 0; blk < NN / TBLK; ++blk) {
    // ---- load X block (16x32 = 512 floats) and y block ----
    {
      const float4* s4 = (const float4*)(Xg + blk * TBLK * DDIM);
      float4*       d4 = (float4*)Xsh;
      #pragma unroll
      for (int idx = lane; idx < (TBLK * DDIM) / 4; idx += 32) d4[idx] = s4[idx];
    }
    if (lane < TBLK) ybs[lane] = Yg[blk * TBLK + lane];
    if (blk + 1 < NN / TBLK)
      __builtin_prefetch(Xg + (blk + 1) * TBLK * DDIM + lane * 16, 0, 3);
    __syncthreads();

    // ---- V = X * P   (16x32, two 16-wide tiles, K=32) ----
    #pragma unroll
    for (int t = 0; t < 2; ++t) {
      v8f c = {};
      #pragma unroll
      for (int kk = 0; kk < DDIM; kk += 4) {
        v2f a, bb;
        a.x  = Xsh[m * DDIM + kk + k2];
        a.y  = Xsh[m * DDIM + kk + k2 + 1];
        bb.x = Psh[(kk + k2) * DDIM + t * 16 + n];
        bb.y = Psh[(kk + k2 + 1) * DDIM + t * 16 + n];
        c = __builtin_amdgcn_wmma_f32_16x16x4_f32(false, a, false, bb,
                                                  (short)0, c, false, false);
      }
      #pragma unroll
      for (int v = 0; v < 8; ++v) Vsh[(v + 8 * g) * DDIM + t * 16 + n] = c[v];
    }
    __syncthreads();

    // ---- S = V * X^T  (16x16, K=32) ----
    {
      v8f c = {};
      #pragma unroll
      for (int kk = 0; kk < DDIM; kk += 4) {
        v2f a, bb;
        a.x  = Vsh[m * DDIM + kk + k2];
        a.y  = Vsh[m * DDIM + kk + k2 + 1];
        bb.x = Xsh[n * DDIM + kk + k2];        // B = X^T : B[k][n] = X[n][k]
        bb.y = Xsh[n * DDIM + kk + k2 + 1];
        c = __builtin_amdgcn_wmma_f32_16x16x4_f32(false, a, false, bb,
                                                  (short)0, c, false, false);
      }
      #pragma unroll
      for (int v = 0; v < 8; ++v) Ssh[(v + 8 * g) * 16 + n] = c[v];
    }
    __syncthreads();

    // ---- v0 = X*w ; C = 0 ----
    if (lane < TBLK) {
      float acc = 0.0f;
      #pragma unroll
      for (int kd = 0; kd < DDIM; ++kd) acc += Xsh[lane * DDIM + kd] * wsh[kd];
      v0s[lane] = acc;
    }
    for (int idx = lane; idx < TBLK * TBLK; idx += 32) Csh[idx] = 0.0f;
    float greg = 0.0f;   // g[lane] for lane < 16
    __syncthreads();

    // ---- 16 exact sequential RLS substeps in compressed space ----
    for (int ms = 0; ms < TBLK; ++ms) {
      float sj = (lane < TBLK) ? Ssh[lane * 16 + ms] : 0.0f;  // s = S[:,ms]
      float aj = 0.0f;
      if (lane < TBLK) {
        float acc = 0.0f;
        #pragma unroll
        for (int l = 0; l < TBLK; ++l) acc += Csh[lane * 16 + l] * Ssh[l * 16 + ms];
        aj = ((lane == ms) ? 1.0f : 0.0f) - acc;              // a = e_ms - C s
      }
      __syncthreads();                 // previous-iter readers of avs done
      if (lane < TBLK) avs[lane] = aj;
      float pd = sj * aj;              // -> s.a
      float pg = sj * greg;            // -> s.g
      #pragma unroll
      for (int off = 8; off >= 1; off >>= 1) {
        pd += __shfl_xor(pd, off, 32);
        pg += __shfl_xor(pg, off, 32);
      }
      __syncthreads();                 // avs visible to all lanes
      float d    = 1.0f + pd;
      float pred = v0s[ms] + pg;       // x_i . w_i  (exact ridge prediction)
      float eps  = ybs[ms] - pred;
      float invd = 1.0f / d;
      if (lane == 0) Og[blk * TBLK + ms] = pred;   // pred at global i=0 is exactly 0
      if (lane < TBLK) {
        greg += aj * eps * invd;                   // g += a*eps/d
        float scale = aj * invd;
        #pragma unroll
        for (int l = 0; l < TBLK; ++l)             // C += a a^T / d (own row only)
          Csh[lane * 16 + l] += scale * avs[l];
      }
    }
    if (lane < TBLK) gvs[lane] = greg;
    __syncthreads();

    // ---- W = C * V   (16x32, K=16) ----
    #pragma unroll
    for (int t = 0; t < 2; ++t) {
      v8f c = {};
      #pragma unroll
      for (int kk = 0; kk < TBLK; kk += 4) {
        v2f a, bb;
        a.x  = Csh[m * 16 + kk + k2];
        a.y  = Csh[m * 16 + kk + k2 + 1];
        bb.x = Vsh[(kk + k2) * DDIM + t * 16 + n];
        bb.y = Vsh[(kk + k2 + 1) * DDIM + t * 16 + n];
        c = __builtin_amdgcn_wmma_f32_16x16x4_f32(false, a, false, bb,
                                                  (short)0, c, false, false);
      }
      #pragma unroll
      for (int v = 0; v < 8; ++v) Wsh[(v + 8 * g) * DDIM + t * 16 + n] = c[v];
    }
    // ---- w += V^T g  (independent of Wsh; all 32 lanes) ----
    {
      float acc = 0.0f;
      #pragma unroll
      for (int l = 0; l < TBLK; ++l) acc += Vsh[l * DDIM + lane] * gvs[l];
      wsh[lane] += acc;
    }
    __syncthreads();

    // ---- P -= V^T * W  (32x32, four 16x16 tiles, K=16) ----
    #pragma unroll
    for (int mt = 0; mt < 2; ++mt) {
      #pragma unroll
      for (int nt = 0; nt < 2; ++nt) {
        v8f c = {};
        #pragma unroll
        for (int kk = 0; kk < TBLK; kk += 4) {
          v2f a, bb;
          a.x  = Vsh[(kk + k2) * DDIM + mt * 16 + m];      // A = V^T
          a.y  = Vsh[(kk + k2 + 1) * DDIM + mt * 16 + m];
          bb.x = Wsh[(kk + k2) * DDIM + nt * 16 + n];
          bb.y = Wsh[(kk + k2 + 1) * DDIM + nt * 16 + n];
          c = __builtin_amdgcn_wmma_f32_16x16x4_f32(false, a, false, bb,
                                                    (short)0, c, false, false);
        }
        #pragma unroll
        for (int v = 0; v < 8; ++v)
          Psh[(mt * 16 + v + 8 * g) * DDIM + nt * 16 + n] -= c[v];
      }
    }
    __syncthreads();
  }
}

extern "C" void kernel_launch(void* const* d_in, const int* in_sizes, int n_in,
                              void* d_out, int out_size, void* d_ws, size_t ws_size,
                              hipStream_t stream) {
  (void)in_sizes; (void)n_in; (void)d_ws; (void)ws_size; (void)out_size;
  const float* data    = (const float*)d_in[0];   // [256,256,32] f32
  const float* targets = (const float*)d_in[1];   // [256,256]    f32
  float*       out     = (float*)d_out;           // [256,256]    f32
  ridge_rls_wmma<<<BB, 32, 0, stream>>>(data, targets, out);
}